// Net_18889266168118
// MI455X (gfx1250) — compile-verified
//
#include <hip/hip_runtime.h>

typedef __attribute__((ext_vector_type(2))) float v2f;
typedef __attribute__((ext_vector_type(4))) float v4f;
typedef __attribute__((ext_vector_type(8))) float v8f;

// Select one of 9 broadcast weights by a lane-varying index without spilling
// to scratch (pure v_cndmask chain; runs once per wave).
__device__ __forceinline__ float pick9(int idx,
    float w0, float w1, float w2, float w3, float w4,
    float w5, float w6, float w7, float w8)
{
    float v = w0;
    v = (idx == 1) ? w1 : v;
    v = (idx == 2) ? w2 : v;
    v = (idx == 3) ? w3 : v;
    v = (idx == 4) ? w4 : v;
    v = (idx == 5) ? w5 : v;
    v = (idx == 6) ? w6 : v;
    v = (idx == 7) ? w7 : v;
    v = (idx == 8) ? w8 : v;
    return v;
}

// Submanifold 3x3 conv on independent 4x4 tiles as Y = M(16x16) * X(16x16),
// where M encodes the SAME-padded stencil and X packs 16 tiles as columns.
// Computed as 4 chained V_WMMA_F32_16X16X4_F32 (K = 16 split into 4).
__global__ __launch_bounds__(256) void subm_conv4x4_wmma(
    const float* __restrict__ x,
    const float* __restrict__ W,
    float* __restrict__ y,
    int nGroups)
{
    const int lane          = threadIdx.x & 31;
    const int wavesPerBlock = blockDim.x >> 5;
    const int waveId        = blockIdx.x * wavesPerBlock + (threadIdx.x >> 5);
    const int nWaves        = gridDim.x * wavesPerBlock;

    const int col  = lane & 15;   // tile column within the 16-tile group
    const int half = lane >> 4;   // 0: lanes 0-15, 1: lanes 16-31

    // Broadcast the 9 stencil weights (uniform address -> scalar loads).
    const float w0 = W[0], w1 = W[1], w2 = W[2];
    const float w3 = W[3], w4 = W[4], w5 = W[5];
    const float w6 = W[6], w7 = W[7], w8 = W[8];

    // Build A fragments once per wave. Chunk k is M[:, 4k..4k+3] (16x4 f32).
    // ISA layout: lanes 0-15 carry M-rows; VGPR0 = K even, VGPR1 = K odd,
    // with lanes 16-31 taking K+2.
    const int m  = col;           // output pixel index (row of M)
    const int oi = m >> 2, oj = m & 3;
    v2f a[4];
#pragma unroll
    for (int k = 0; k < 4; ++k) {
        float av[2];
#pragma unroll
        for (int e = 0; e < 2; ++e) {
            const int q  = 4 * k + 2 * half + e;   // input pixel index (col of M)
            const int r  = q >> 2, c = q & 3;
            const int dr = r - oi + 1, dc = c - oj + 1;
            const bool inb = (dr >= 0) & (dr < 3) & (dc >= 0) & (dc < 3);
            const float wv = pick9(dr * 3 + dc, w0, w1, w2, w3, w4, w5, w6, w7, w8);
            av[e] = inb ? wv : 0.0f;
        }
        a[k].x = av[0];
        a[k].y = av[1];
    }

    // Grid-stride over 16-tile groups (1 KB in / 1 KB out per wave-iteration).
    for (int g = waveId; g < nGroups; g += nWaves) {
        const float* xg = x + (size_t)g * 256;
        float*       yg = y + (size_t)g * 256;

        // Pull the next group's cachelines early (global_prefetch_b8).
        if (g + nWaves < nGroups) {
            __builtin_prefetch(x + (size_t)(g + nWaves) * 256 + lane * 8, 0, 0);
        }

        // B fragments: chunk k needs X[4k + 2*half + {0,1}][col] -> b64 loads.
        v2f b[4];
#pragma unroll
        for (int k = 0; k < 4; ++k)
            b[k] = *(const v2f*)(xg + col * 16 + 4 * k + 2 * half);

        // Inputs at this lane's output positions (activity mask) -> b128 loads.
        const float* xmp = xg + col * 16 + half * 8;
        const v4f xlo = *(const v4f*)(xmp);
        const v4f xhi = *(const v4f*)(xmp + 4);

        v8f acc = {};
        acc = __builtin_amdgcn_wmma_f32_16x16x4_f32(false, a[0], false, b[0], (short)0, acc, false, false);
        acc = __builtin_amdgcn_wmma_f32_16x16x4_f32(false, a[1], false, b[1], (short)0, acc, false, false);
        acc = __builtin_amdgcn_wmma_f32_16x16x4_f32(false, a[2], false, b[2], (short)0, acc, false, false);
        acc = __builtin_amdgcn_wmma_f32_16x16x4_f32(false, a[3], false, b[3], (short)0, acc, false, false);

        // Submanifold mask: zero outputs where the input pixel was zero.
        v4f o0, o1;
#pragma unroll
        for (int v = 0; v < 4; ++v) {
            o0[v] = (xlo[v] != 0.0f) ? acc[v]     : 0.0f;
            o1[v] = (xhi[v] != 0.0f) ? acc[v + 4] : 0.0f;
        }

        float* yp = yg + col * 16 + half * 8;
        *(v4f*)(yp)     = o0;   // pixels half*8 + 0..3
        *(v4f*)(yp + 4) = o1;   // pixels half*8 + 4..7
    }
}

extern "C" void kernel_launch(void* const* d_in, const int* in_sizes, int n_in,
                              void* d_out, int out_size, void* d_ws, size_t ws_size,
                              hipStream_t stream)
{
    const float* x = (const float*)d_in[0];   // [N,4,4,1] f32
    const float* W = (const float*)d_in[1];   // [3,3,1,1] f32
    float*       y = (float*)d_out;           // [N,1,4,4] f32 (same flat layout)

    int nGroups = in_sizes[0] / 256;          // 16 tiles of 16 floats per group
    if (nGroups < 1) nGroups = 1;

    // ~8 groups per wave to amortize the A-fragment build; 8 waves per block.
    int wavesWanted = (nGroups + 7) / 8;
    int blocks = (wavesWanted + 7) / 8;
    if (blocks < 1) blocks = 1;
    if (blocks > 16384) blocks = 16384;

    subm_conv4x4_wmma<<<blocks, 256, 0, stream>>>(x, W, y, nGroups);
}